// BipartiteCitationGNN_87247965651651
// MI455X (gfx1250) — compile-verified
//
#include <hip/hip_runtime.h>

typedef __attribute__((ext_vector_type(2))) float v2f;
typedef __attribute__((ext_vector_type(8))) float v8f;

#define HDIM 64

// ---------------------------------------------------------------------------
// FP32 WMMA micro-op: D(16x16) = A(16x4) * B(4x16) + C   (V_WMMA_F32_16X16X4_F32)
// ---------------------------------------------------------------------------
__device__ __forceinline__ v8f wmma4(v2f a, v2f b, v8f c) {
  return __builtin_amdgcn_wmma_f32_16x16x4_f32(
      /*neg_a=*/false, a, /*neg_b=*/false, b,
      /*c_mod=*/(short)0, c, /*reuse_a=*/false, /*reuse_b=*/false);
}

// ---------------------------------------------------------------------------
// Non-returning HW float atomic add (GLOBAL_ATOMIC_ADD_F32, STOREcnt-tracked).
// Inline asm guarantees no CAS-loop fallback and no pre-op value return.
// Cross-kernel ordering is provided by stream serialization + the implicit
// S_WAIT_IDLE of S_ENDPGM.
// ---------------------------------------------------------------------------
__device__ __forceinline__ void fatomic_add(float* p, float v) {
  asm volatile("global_atomic_add_f32 %0, %1, off"
               :
               : "v"(p), "v"(v)
               : "memory");
}

// ---------------------------------------------------------------------------
// Zero a float region (grid-stride)
// ---------------------------------------------------------------------------
__global__ __launch_bounds__(256) void zero_f32(float* __restrict__ p, long long n) {
  long long i = (long long)blockIdx.x * blockDim.x + threadIdx.x;
  long long stride = (long long)gridDim.x * blockDim.x;
  for (; i < n; i += stride) p[i] = 0.0f;
}

// ---------------------------------------------------------------------------
// Per-edge degree counts (fp32 atomics; clip-to-1 is applied at consumption)
// ---------------------------------------------------------------------------
__global__ __launch_bounds__(256) void degrees_kernel(
    const int* __restrict__ esrc, const int* __restrict__ edst,
    float* __restrict__ cnt_a, float* __restrict__ cnt_p, int nedges) {
  int e = blockIdx.x * blockDim.x + threadIdx.x;
  if (e >= nedges) return;
  fatomic_add(&cnt_a[esrc[e]], 1.0f);
  fatomic_add(&cnt_p[edst[e]], 1.0f);
}

// ---------------------------------------------------------------------------
// Fused bidirectional scatter-add: aggP[dst] += a0[src]; aggA[src] += p0[dst]
// 16 threads per edge, float4 per thread (coalesced row gathers).
// ---------------------------------------------------------------------------
__global__ __launch_bounds__(256) void scatter_both(
    const float* __restrict__ a0, const float* __restrict__ p0,
    float* __restrict__ aggP, float* __restrict__ aggA,
    const int* __restrict__ esrc, const int* __restrict__ edst, int nedges) {
  long long tid = (long long)blockIdx.x * blockDim.x + threadIdx.x;
  long long total = (long long)nedges * 16;
  if (tid >= total) return;
  int e = (int)(tid >> 4);
  int q4 = (int)(tid & 15) * 4;
  int src = esrc[e];
  int dst = edst[e];
  float4 va = *(const float4*)(a0 + (long long)src * HDIM + q4);
  float4 vp = *(const float4*)(p0 + (long long)dst * HDIM + q4);
  float* dp = aggP + (long long)dst * HDIM + q4;
  float* da = aggA + (long long)src * HDIM + q4;
  fatomic_add(dp + 0, va.x); fatomic_add(dp + 1, va.y);
  fatomic_add(dp + 2, va.z); fatomic_add(dp + 3, va.w);
  fatomic_add(da + 0, vp.x); fatomic_add(da + 1, vp.y);
  fatomic_add(da + 2, vp.z); fatomic_add(da + 3, vp.w);
}

// One-direction scatter: aggP[dst] += x[src]
__global__ __launch_bounds__(256) void scatter_one(
    const float* __restrict__ x, float* __restrict__ aggP,
    const int* __restrict__ esrc, const int* __restrict__ edst, int nedges) {
  long long tid = (long long)blockIdx.x * blockDim.x + threadIdx.x;
  long long total = (long long)nedges * 16;
  if (tid >= total) return;
  int e = (int)(tid >> 4);
  int q4 = (int)(tid & 15) * 4;
  int src = esrc[e];
  int dst = edst[e];
  float4 v = *(const float4*)(x + (long long)src * HDIM + q4);
  float* dp = aggP + (long long)dst * HDIM + q4;
  fatomic_add(dp + 0, v.x); fatomic_add(dp + 1, v.y);
  fatomic_add(dp + 2, v.z); fatomic_add(dp + 3, v.w);
}

// ---------------------------------------------------------------------------
// Templated WMMA GEMM:
//   Y = [relu]( (A1 [*1/max(cnt,1)]) @ W1m  [+ A2 @ W2m]  + bias )
// A1: [nrows,K1], W1m: [K1,64], A2: [nrows,64], W2m: [64,64], Y: [nrows,64]
// One wave computes a 16x64 tile (4 accumulators). 8 waves / block.
// ---------------------------------------------------------------------------
template <bool SCALE, bool DUAL, bool RELU>
__global__ __launch_bounds__(256) void sage_gemm(
    const float* __restrict__ A1, const float* __restrict__ W1m,
    const float* __restrict__ cnt,
    const float* __restrict__ A2, const float* __restrict__ W2m,
    const float* __restrict__ bias, float* __restrict__ Y,
    int nrows, int K1) {
  const int lane = threadIdx.x & 31;
  const int wave = threadIdx.x >> 5;
  const int tile = blockIdx.x * 8 + wave;
  if (tile * 16 >= nrows) return;          // wave-uniform: EXEC stays all-1s
  const int rowbase = tile * 16;
  const int half = lane >> 4;              // lanes 16..31 carry K+2 (A) / K+2 rows (B)
  const int l16 = lane & 15;
  const int rowA = rowbase + l16;          // A-matrix: lane%16 = M
  const int koff = half * 2;

  float inv = 1.0f;
  if (SCALE) {
    float cv = cnt[rowA];
    inv = 1.0f / (cv > 1.0f ? cv : 1.0f);
  }

  v8f c[4];
  const v8f vzero = {0.f, 0.f, 0.f, 0.f, 0.f, 0.f, 0.f, 0.f};
#pragma unroll
  for (int nt = 0; nt < 4; ++nt) c[nt] = vzero;

  // --- A1 @ W1m ---
  {
    const float* Arow = A1 + (long long)rowA * K1;
    for (int k = 0; k < K1; k += 4) {
      v2f a;
      a.x = Arow[k + koff];
      a.y = Arow[k + koff + 1];
      if (SCALE) { a.x *= inv; a.y *= inv; }
#pragma unroll
      for (int nt = 0; nt < 4; ++nt) {
        const int col = nt * 16 + l16;     // B-matrix: lane%16 = N
        v2f b;
        b.x = W1m[(k + koff) * HDIM + col];
        b.y = W1m[(k + koff + 1) * HDIM + col];
        c[nt] = wmma4(a, b, c[nt]);
      }
    }
  }

  // --- + A2 @ W2m ---
  if (DUAL) {
    const float* Xrow = A2 + (long long)rowA * HDIM;
    for (int k = 0; k < HDIM; k += 4) {
      v2f a;
      a.x = Xrow[k + koff];
      a.y = Xrow[k + koff + 1];
#pragma unroll
      for (int nt = 0; nt < 4; ++nt) {
        const int col = nt * 16 + l16;
        v2f b;
        b.x = W2m[(k + koff) * HDIM + col];
        b.y = W2m[(k + koff + 1) * HDIM + col];
        c[nt] = wmma4(a, b, c[nt]);
      }
    }
  }

  // --- epilogue: bias (+relu), store. C/D layout: vgpr r -> row r (+8 for upper half)
#pragma unroll
  for (int nt = 0; nt < 4; ++nt) {
    const int col = nt * 16 + l16;
    const float bv = bias[col];
#pragma unroll
    for (int r = 0; r < 8; ++r) {
      const int row = rowbase + r + 8 * half;
      float v = c[nt][r] + bv;
      if (RELU) v = v > 0.f ? v : 0.f;
      Y[(long long)row * HDIM + col] = v;
    }
  }
}

// ---------------------------------------------------------------------------
// Head: out[row] = relu(X@W1 + b1) @ W2 + b2
// WMMA for X@W1, then per-row dot with W2 done in C/D layout via shfl_xor
// reduction across each 16-lane half of the wave32.
// ---------------------------------------------------------------------------
__global__ __launch_bounds__(256) void mlp_head(
    const float* __restrict__ X, const float* __restrict__ W1,
    const float* __restrict__ b1, const float* __restrict__ W2,
    const float* __restrict__ b2, float* __restrict__ out, int nrows) {
  const int lane = threadIdx.x & 31;
  const int wave = threadIdx.x >> 5;
  const int tile = blockIdx.x * 8 + wave;
  if (tile * 16 >= nrows) return;
  const int rowbase = tile * 16;
  const int half = lane >> 4;
  const int l16 = lane & 15;
  const int rowA = rowbase + l16;
  const int koff = half * 2;

  v8f c[4];
  const v8f vzero = {0.f, 0.f, 0.f, 0.f, 0.f, 0.f, 0.f, 0.f};
#pragma unroll
  for (int nt = 0; nt < 4; ++nt) c[nt] = vzero;

  const float* Arow = X + (long long)rowA * HDIM;
  for (int k = 0; k < HDIM; k += 4) {
    v2f a;
    a.x = Arow[k + koff];
    a.y = Arow[k + koff + 1];
#pragma unroll
    for (int nt = 0; nt < 4; ++nt) {
      const int col = nt * 16 + l16;
      v2f b;
      b.x = W1[(k + koff) * HDIM + col];
      b.y = W1[(k + koff + 1) * HDIM + col];
      c[nt] = wmma4(a, b, c[nt]);
    }
  }

  float w2v[4], b1v[4];
#pragma unroll
  for (int nt = 0; nt < 4; ++nt) {
    w2v[nt] = W2[nt * 16 + l16];
    b1v[nt] = b1[nt * 16 + l16];
  }
  const float b2v = b2[0];

#pragma unroll
  for (int r = 0; r < 8; ++r) {
    float acc = 0.f;
#pragma unroll
    for (int nt = 0; nt < 4; ++nt) {
      float h = c[nt][r] + b1v[nt];
      h = h > 0.f ? h : 0.f;
      acc += h * w2v[nt];
    }
    // reduce over the 16 lanes of this half-wave (masks < 16 stay in-group)
    acc += __shfl_xor(acc, 1, 32);
    acc += __shfl_xor(acc, 2, 32);
    acc += __shfl_xor(acc, 4, 32);
    acc += __shfl_xor(acc, 8, 32);
    if (l16 == 0) out[rowbase + r + 8 * half] = acc + b2v;
  }
}

// ---------------------------------------------------------------------------
// Host-side orchestration
// ---------------------------------------------------------------------------
extern "C" void kernel_launch(void* const* d_in, const int* in_sizes, int n_in,
                              void* d_out, int out_size, void* d_ws, size_t ws_size,
                              hipStream_t stream) {
  const int CIN = 128;
  const int NA = in_sizes[0] / CIN;   // 100000
  const int NP = in_sizes[1] / HDIM;  // 300000
  const int E  = in_sizes[20];        // 1000000

  const float* author = (const float*)d_in[0];
  const float* p0     = (const float*)d_in[1];
  const float* Wproj  = (const float*)d_in[2];
  const float* bproj  = (const float*)d_in[3];
  const float* c1p_Wl = (const float*)d_in[4];
  const float* c1p_bl = (const float*)d_in[5];
  const float* c1p_Wr = (const float*)d_in[6];
  const float* c1a_Wl = (const float*)d_in[7];
  const float* c1a_bl = (const float*)d_in[8];
  const float* c1a_Wr = (const float*)d_in[9];
  const float* c2p_Wl = (const float*)d_in[10];
  const float* c2p_bl = (const float*)d_in[11];
  const float* c2p_Wr = (const float*)d_in[12];
  // d_in[13..15] = c2a_* : dead code in the reference (author output of conv2 unused)
  const float* W1 = (const float*)d_in[16];
  const float* b1 = (const float*)d_in[17];
  const float* W2 = (const float*)d_in[18];
  const float* b2 = (const float*)d_in[19];
  const int* esrc = (const int*)d_in[20];
  const int* edst = (const int*)d_in[21];

  // Workspace layout (floats). aggP is reused for both paper aggregation passes.
  float* ws = (float*)d_ws;
  float* cnt_p = ws;                                  // NP
  float* cnt_a = cnt_p + NP;                          // NA
  float* aggP  = cnt_a + NA;                          // NP*64
  float* aggA  = aggP + (long long)NP * HDIM;         // NA*64
  float* p1    = aggA + (long long)NA * HDIM;         // NP*64
  float* a0    = p1   + (long long)NP * HDIM;         // NA*64
  float* a1    = a0   + (long long)NA * HDIM;         // NA*64
  float* p2    = a1   + (long long)NA * HDIM;         // NP*64
  (void)ws_size; (void)n_in; (void)out_size;

  const long long scatterThreads = (long long)E * 16;
  const int scatterBlocks = (int)((scatterThreads + 255) / 256);
  const int tilesP = (NP + 15) / 16, blocksP = (tilesP + 7) / 8;
  const int tilesA = (NA + 15) / 16, blocksA = (tilesA + 7) / 8;

  // 1) zero cnt_p, cnt_a, aggP, aggA (contiguous region)
  const long long zcount =
      (long long)NP + NA + (long long)NP * HDIM + (long long)NA * HDIM;
  zero_f32<<<2048, 256, 0, stream>>>(ws, zcount);

  // 2) degrees
  degrees_kernel<<<(E + 255) / 256, 256, 0, stream>>>(esrc, edst, cnt_a, cnt_p, E);

  // 3) a0 = author_features @ Wproj + bproj   (K = 128)
  sage_gemm<false, false, false><<<blocksA, 256, 0, stream>>>(
      author, Wproj, nullptr, nullptr, nullptr, bproj, a0, NA, CIN);

  // 4) aggP[dst] += a0[src]; aggA[src] += p0[dst]
  scatter_both<<<scatterBlocks, 256, 0, stream>>>(a0, p0, aggP, aggA, esrc, edst, E);

  // 5) p1 = relu((aggP/cnt_p) @ c1p_Wl + c1p_bl + p0 @ c1p_Wr)
  sage_gemm<true, true, true><<<blocksP, 256, 0, stream>>>(
      aggP, c1p_Wl, cnt_p, p0, c1p_Wr, c1p_bl, p1, NP, HDIM);

  // 6) a1 = relu((aggA/cnt_a) @ c1a_Wl + c1a_bl + a0 @ c1a_Wr)
  sage_gemm<true, true, true><<<blocksA, 256, 0, stream>>>(
      aggA, c1a_Wl, cnt_a, a0, c1a_Wr, c1a_bl, a1, NA, HDIM);

  // 7) re-zero aggP, then aggP[dst] += a1[src]
  zero_f32<<<2048, 256, 0, stream>>>(aggP, (long long)NP * HDIM);
  scatter_one<<<scatterBlocks, 256, 0, stream>>>(a1, aggP, esrc, edst, E);

  // 8) p2 = relu((aggP/cnt_p) @ c2p_Wl + c2p_bl + p1 @ c2p_Wr)
  sage_gemm<true, true, true><<<blocksP, 256, 0, stream>>>(
      aggP, c2p_Wl, cnt_p, p1, c2p_Wr, c2p_bl, p2, NP, HDIM);

  // 9) out = relu(p2 @ W1 + b1) @ W2 + b2
  mlp_head<<<blocksP, 256, 0, stream>>>(p2, W1, b1, W2, b2, (float*)d_out, NP);
}